// LSSImVoxelViewTransform_10264971837634
// MI455X (gfx1250) — compile-verified
//
#include <hip/hip_runtime.h>
#include <hip/hip_bf16.h>

typedef float v2f __attribute__((ext_vector_type(2)));
typedef float v8f __attribute__((ext_vector_type(8)));

#define N_CAM   6
#define C_IN    64
#define DBINS   66
#define O_TOT   130      // 66 depth bins + 64 feature channels
#define IMH     32
#define IMW     88
#define PIX     2816     // 32*88
#define NPTS    131072   // 128*128*8
#define O_TILES 9        // ceil(130/16)
#define M_TILES 176      // 2816/16

// ---------------------------------------------------------------------------
// Kernel A: fused 1x1 conv (WMMA fp32) + depth softmax.
//   depth: (6, 66, 2816)   feat: (6, 64, 2816)   both fp32 in workspace.
// One wave handles 16 pixels (N dim) x all 130 output channels (M dim).
// ---------------------------------------------------------------------------
__global__ __launch_bounds__(128) void dn_softmax_kernel(
    const float* __restrict__ x_fov, const float* __restrict__ w_dn,
    const float* __restrict__ b_dn, float* __restrict__ depth,
    float* __restrict__ feat)
{
    const int wave = blockIdx.x * 4 + (threadIdx.x >> 5);
    const int lane = threadIdx.x & 31;
    const int half = lane >> 4;      // 0: rows 0-7 / K 0-1 ; 1: rows 8-15 / K 2-3
    const int li   = lane & 15;      // pixel column within tile (N), also A row (M)
    const int n    = wave / M_TILES;
    const int pb   = (wave % M_TILES) * 16;

    const float* xn = x_fov + n * (C_IN * PIX);

    v8f acc[O_TILES] = {};

    // GEMM: D(o,pixel) += A(o,c) * B(c,pixel), K=64 in 16 steps of 4.
    for (int kb = 0; kb < 16; ++kb) {
        const int k0 = kb * 4 + half * 2;
        v2f bm;
        bm.x = xn[(k0    ) * PIX + pb + li];
        bm.y = xn[(k0 + 1) * PIX + pb + li];
        #pragma unroll
        for (int t = 0; t < O_TILES; ++t) {
            int row = 16 * t + li;
            row = row > (O_TOT - 1) ? (O_TOT - 1) : row;   // pad tile 8 safely
            v2f am = *(const v2f*)(w_dn + row * C_IN + k0);
            acc[t] = __builtin_amdgcn_wmma_f32_16x16x4_f32(
                false, am, false, bm, (short)0, acc[t], false, false);
        }
    }

    // Bias add (acc[t][r] holds o = 16t + 8*half + r for pixel pb+li).
    #pragma unroll
    for (int t = 0; t < O_TILES; ++t) {
        #pragma unroll
        for (int r = 0; r < 8; ++r) {
            int o = 16 * t + 8 * half + r;
            o = o > (O_TOT - 1) ? (O_TOT - 1) : o;
            acc[t][r] += b_dn[o];
        }
    }

    // Softmax over depth bins o < 66. Rows for one pixel live in lane l and l^16.
    float m = -3.0e38f;
    #pragma unroll
    for (int t = 0; t < 4; ++t)
        #pragma unroll
        for (int r = 0; r < 8; ++r) m = fmaxf(m, acc[t][r]);
    if (half == 0) { m = fmaxf(m, acc[4][0]); m = fmaxf(m, acc[4][1]); }
    m = fmaxf(m, __shfl_xor(m, 16, 32));

    float s = 0.0f;
    #pragma unroll
    for (int t = 0; t < 4; ++t)
        #pragma unroll
        for (int r = 0; r < 8; ++r) s += __expf(acc[t][r] - m);
    if (half == 0) { s += __expf(acc[4][0] - m) + __expf(acc[4][1] - m); }
    s += __shfl_xor(s, 16, 32);
    const float inv = 1.0f / s;

    // Write depth (d-major so kernel B's wave-uniform gathers broadcast in L1).
    float* dpx = depth + n * (DBINS * PIX) + pb + li;
    #pragma unroll
    for (int t = 0; t < 4; ++t)
        #pragma unroll
        for (int r = 0; r < 8; ++r) {
            int d = 16 * t + 8 * half + r;
            dpx[d * PIX] = __expf(acc[t][r] - m) * inv;
        }
    if (half == 0) {
        dpx[64 * PIX] = __expf(acc[4][0] - m) * inv;
        dpx[65 * PIX] = __expf(acc[4][1] - m) * inv;
    }

    // Write feat channels (o >= 66).
    float* fpx = feat + n * (C_IN * PIX) + pb + li;
    #pragma unroll
    for (int t = 4; t < O_TILES; ++t)
        #pragma unroll
        for (int r = 0; r < 8; ++r) {
            int o = 16 * t + 8 * half + r;
            int c = o - DBINS;
            if (c >= 0 && c < C_IN) fpx[c * PIX] = acc[t][r];
        }
}

// ---------------------------------------------------------------------------
// Kernel B: per-voxel projection + factored trilinear gather + camera average.
// One wave per point; lane covers channels (lane, lane+32).
// Projection/depth work is wave-uniform -> broadcast loads; feat gathers are
// per-lane, L2-resident (whole feat tensor is 4.3 MB << 192 MB L2).
// ---------------------------------------------------------------------------
__global__ __launch_bounds__(256) void splat_kernel(
    const float* __restrict__ depth, const float* __restrict__ feat,
    const float* __restrict__ points, const float* __restrict__ proj,
    float* __restrict__ out)
{
    const int p    = blockIdx.x * 8 + (threadIdx.x >> 5);
    const int lane = threadIdx.x & 31;

    const float px = points[p * 3 + 0];
    const float py = points[p * 3 + 1];
    const float pz = points[p * 3 + 2];

    float acc0 = 0.0f, acc1 = 0.0f;
    int cnt = 0;

    for (int nc = 0; nc < N_CAM; ++nc) {
        const float* P = proj + nc * 16;
        const float X = P[0] * px + P[1] * py + P[2]  * pz + P[3];
        const float Y = P[4] * px + P[5] * py + P[6]  * pz + P[7];
        const float Z = P[8] * px + P[9] * py + P[10] * pz + P[11];
        const float zs = fabsf(Z) > 0.001f ? Z : 0.001f;
        const float u = X / zs;
        const float v = Y / zs;
        if (!(u >= 0.0f && u < 1408.0f && v >= 0.0f && v < 512.0f &&
              Z > 0.0f && Z < 50.0f))
            continue;
        ++cnt;

        const float gx = u * ((float)IMW / 1408.0f) - 0.5f;
        const float gy = v * ((float)IMH / 512.0f)  - 0.5f;
        const float gz = Z * ((float)DBINS / 50.0f) - 0.5f;
        const float x0f = floorf(gx), y0f = floorf(gy), z0f = floorf(gz);
        const int   x0 = (int)x0f,    y0 = (int)y0f,    z0 = (int)z0f;
        const float tx = gx - x0f,    ty = gy - y0f,    tz = gz - z0f;

        const float wz0 = (z0     >= 0 && z0     < DBINS) ? (1.0f - tz) : 0.0f;
        const float wz1 = (z0 + 1 >= 0 && z0 + 1 < DBINS) ? tz          : 0.0f;
        const int z0c = min(max(z0,     0), DBINS - 1);
        const int z1c = min(max(z0 + 1, 0), DBINS - 1);

        const float* dN = depth + nc * (DBINS * PIX);
        const float* fN = feat  + nc * (C_IN  * PIX);

        #pragma unroll
        for (int dy = 0; dy < 2; ++dy) {
            const int yi = y0 + dy;
            if (yi < 0 || yi >= IMH) continue;
            const float wy = dy ? ty : 1.0f - ty;
            #pragma unroll
            for (int dx = 0; dx < 2; ++dx) {
                const int xi = x0 + dx;
                if (xi < 0 || xi >= IMW) continue;
                const float wx = dx ? tx : 1.0f - tx;
                const int pix = yi * IMW + xi;
                // depth interpolated over z, shared by all 64 channels
                const float dv = wz0 * dN[z0c * PIX + pix] +
                                 wz1 * dN[z1c * PIX + pix];
                const float a = wx * wy * dv;
                acc0 += a * fN[(lane     ) * PIX + pix];
                acc1 += a * fN[(lane + 32) * PIX + pix];
            }
        }
    }

    const float scale = cnt > 0 ? 1.0f / (float)cnt : 0.0f;
    const int ix = p & 127;
    const int iy = (p >> 7) & 127;
    const int iz = p >> 14;
    const int oidx = ix * (128 * 8) + iy * 8 + iz;   // out is (C, nx, ny, nz)
    out[(lane     ) * NPTS + oidx] = acc0 * scale;
    out[(lane + 32) * NPTS + oidx] = acc1 * scale;
}

extern "C" void kernel_launch(void* const* d_in, const int* in_sizes, int n_in,
                              void* d_out, int out_size, void* d_ws, size_t ws_size,
                              hipStream_t stream) {
    const float* x_fov  = (const float*)d_in[0];
    const float* w_dn   = (const float*)d_in[1];
    const float* b_dn   = (const float*)d_in[2];
    const float* points = (const float*)d_in[3];
    const float* proj   = (const float*)d_in[4];
    float* out = (float*)d_out;

    float* depth = (float*)d_ws;                        // 6*66*2816 floats
    float* feat  = depth + N_CAM * DBINS * PIX;         // 6*64*2816 floats

    // 6 cams * 176 pixel-tiles = 1056 waves = 264 blocks * 4 waves
    dn_softmax_kernel<<<264, 128, 0, stream>>>(x_fov, w_dn, b_dn, depth, feat);
    // 131072 points, one wave each, 8 waves per block
    splat_kernel<<<NPTS / 8, 256, 0, stream>>>(depth, feat, points, proj, out);
}